// MixLoss_56221121904930
// MI455X (gfx1250) — compile-verified
//
#include <hip/hip_runtime.h>
#include <math.h>

// Volume geometry (B=2, C=1, D=128, H=192, W=192), f32
#define DD 128
#define HH 192
#define WW 192
#define SD_ (HH * WW)                 // 36864
#define SB_ ((size_t)DD * SD_)        // 4718592 per batch entry
#define NV_ ((size_t)2 * SB_)         // 9437184 total elements per tensor
#define NBLK 1024                     // reduction blocks (multiple of 4 for WMMA chunks)

// ---------------- CDNA5 feature probes ----------------
#if defined(__gfx1250__) && __has_builtin(__builtin_amdgcn_global_load_async_to_lds_b32)
#define HAVE_ASYNC_LDS 1
#else
#define HAVE_ASYNC_LDS 0
#endif

typedef __attribute__((address_space(1))) int as1_int;
typedef __attribute__((address_space(3))) int as3_int;

__device__ __forceinline__ void wait_async_then_barrier() {
#if HAVE_ASYNC_LDS
#if __has_builtin(__builtin_amdgcn_s_wait_asynccnt)
  __builtin_amdgcn_s_wait_asynccnt(0);
#else
  asm volatile("s_wait_asynccnt 0" ::: "memory");
#endif
#endif
  __syncthreads();
}

// =====================================================================
// Kernel 1: one soft_skeletonize iteration, fused (cross-min + 3^3 max)
//   x' = relu(x - relu(maxpool3x3x3(mincross(x)) - mincross(x)))
// Tile: 16^3 outputs; x tile 20^3 in LDS (halo 2, +inf padded);
// min field 18^3 in LDS (-inf at out-of-range positions).
// =====================================================================
__global__ __launch_bounds__(256) void skel_iter(const float* __restrict__ src,
                                                 float* __restrict__ dst) {
  __shared__ float xs[20 * 20 * 20];   // 32.0 KB
  __shared__ float mp[18 * 18 * 18];   // 23.3 KB
  const int tid = threadIdx.x;
  const int gx0 = blockIdx.x << 4;
  const int gy0 = blockIdx.y << 4;
  const int b   = blockIdx.z >> 3;          // batch index (0..1)
  const int gz0 = (blockIdx.z & 7) << 4;    // depth tile
  const float* sb = src + (size_t)b * SB_;
  const float INF = __builtin_inff();

  // ---- stage 0: cooperative load of 20^3 x-tile (global offsets -2..17) ----
  for (int idx = tid; idx < 20 * 20 * 20; idx += 256) {
    int iz = idx / 400;
    int r  = idx - iz * 400;
    int iy = r / 20;
    int ix = r - iy * 20;
    int gz = gz0 + iz - 2, gy = gy0 + iy - 2, gx = gx0 + ix - 2;
    bool in = ((unsigned)gz < DD) & ((unsigned)gy < HH) & ((unsigned)gx < WW);
#if HAVE_ASYNC_LDS
    if (in) {
      __builtin_amdgcn_global_load_async_to_lds_b32(
          (as1_int*)(sb + (size_t)gz * SD_ + gy * WW + gx),
          (as3_int*)(&xs[idx]), 0, 0);
    } else {
      xs[idx] = INF;  // +inf: never wins a min (matches -inf padding of maxpool(-x))
    }
#else
    xs[idx] = in ? sb[(size_t)gz * SD_ + gy * WW + gx] : INF;
#endif
  }
  wait_async_then_barrier();

  // ---- stage 1: cross-min field on 18^3 (global offsets -1..16) ----
  for (int c = tid; c < 18 * 18; c += 256) {
    int mx = c % 18, my = c / 18;
    int gy = gy0 + my - 1, gx = gx0 + mx - 1;
    bool xyin = ((unsigned)gy < HH) & ((unsigned)gx < WW);
    int ix = mx + 1, iy = my + 1;
#pragma unroll
    for (int mz = 0; mz < 18; ++mz) {
      int gz = gz0 + mz - 1;
      float v = -INF;  // out-of-range: ignored by the 3^3 max (-inf padding)
      if (xyin && ((unsigned)gz < DD)) {
        int iz = mz + 1;
        float c0 = xs[(iz * 20 + iy) * 20 + ix];
        float z1 = xs[((iz - 1) * 20 + iy) * 20 + ix];
        float z2 = xs[((iz + 1) * 20 + iy) * 20 + ix];
        float y1 = xs[(iz * 20 + (iy - 1)) * 20 + ix];
        float y2 = xs[(iz * 20 + (iy + 1)) * 20 + ix];
        float x1 = xs[(iz * 20 + iy) * 20 + ix - 1];
        float x2 = xs[(iz * 20 + iy) * 20 + ix + 1];
        v = fminf(fminf(fminf(c0, z1), fminf(z2, y1)), fminf(fminf(y2, x1), x2));
      }
      mp[(mz * 18 + my) * 18 + mx] = v;
    }
  }
  __syncthreads();

  // ---- stage 2: per-thread z-column, 2D plane-max once per plane (regs) ----
  const int lx = tid & 15, ly = tid >> 4;
  float pm[18];
#pragma unroll
  for (int mz = 0; mz < 18; ++mz) {
    float m = -INF;
#pragma unroll
    for (int dy = 0; dy < 3; ++dy) {
      int base = (mz * 18 + (ly + dy)) * 18 + lx;
      m = fmaxf(m, fmaxf(fmaxf(mp[base], mp[base + 1]), mp[base + 2]));
    }
    pm[mz] = m;
  }
  float* d = dst + (size_t)b * SB_ + (size_t)gz0 * SD_ + (size_t)(gy0 + ly) * WW + (gx0 + lx);
#pragma unroll
  for (int z = 0; z < 16; ++z) {
    float maxp = fmaxf(pm[z], fmaxf(pm[z + 1], pm[z + 2]));
    float minp = mp[((z + 1) * 18 + (ly + 1)) * 18 + (lx + 1)];
    float xc   = xs[((z + 2) * 20 + (ly + 2)) * 20 + (lx + 2)];
    float contour = fmaxf(maxp - minp, 0.0f);
    d[(size_t)z * SD_] = fmaxf(xc - contour, 0.0f);
  }
}

// =====================================================================
// Kernel 2: fused reductions. 8 sums:
//  0: sum(tskel*pred) 1: sum(tskel) 2: sum(pskel*target) 3: sum(pskel)
//  4: sum(pred*target) 5: sum(pred) 6: sum(target) 7: sum(focal term)
// Deterministic tree reduction (no float atomics).
// =====================================================================
__global__ __launch_bounds__(256) void reduce8(const float* __restrict__ pred,
                                               const float* __restrict__ target,
                                               const float* __restrict__ tskel,
                                               const float* __restrict__ pskel,
                                               float* __restrict__ partials) {
  float s[8] = {0.f, 0.f, 0.f, 0.f, 0.f, 0.f, 0.f, 0.f};
  const size_t stride = (size_t)NBLK * 256;
  for (size_t i = (size_t)blockIdx.x * 256 + threadIdx.x; i < NV_; i += stride) {
    float p = pred[i], t = target[i], ts = tskel[i], ps = pskel[i];
    s[0] += ts * p;
    s[1] += ts;
    s[2] += ps * t;
    s[3] += ps;
    s[4] += p * t;
    s[5] += p;
    s[6] += t;
    // focal (alpha=1, gamma=2), torch BCE log clamp at -100
    float lp  = fmaxf(logf(p), -100.0f);
    float l1p = fmaxf(log1pf(-p), -100.0f);
    float bce = -(t * lp + (1.0f - t) * l1p);
    float pt  = expf(-bce);
    float om  = 1.0f - pt;
    s[7] += om * om * bce;
  }
  // wave32 xor-shuffle reduce
#pragma unroll
  for (int j = 0; j < 8; ++j)
#pragma unroll
    for (int off = 16; off > 0; off >>= 1) s[j] += __shfl_xor(s[j], off, 32);

  __shared__ float wpart[8][8];  // [wave][quantity]
  const int lane = threadIdx.x & 31, wave = threadIdx.x >> 5;
  if (lane == 0) {
#pragma unroll
    for (int j = 0; j < 8; ++j) wpart[wave][j] = s[j];
  }
  __syncthreads();
  if (threadIdx.x < 8) {
    float tot = 0.f;
#pragma unroll
    for (int w = 0; w < 8; ++w) tot += wpart[w][threadIdx.x];
    partials[(size_t)threadIdx.x * NBLK + blockIdx.x] = tot;  // quantity-major
  }
}

// =====================================================================
// Kernel 3: final reduce of NBLK partials per quantity via
// V_WMMA_F32_16X16X4_F32 with B = ones (exact f32 MACs), then epilogue.
// A rows = quantities (8 live rows of 16); accumulate NBLK/4 chunks.
// D layout: VGPR r, lanes 0-15 hold M=r -> lane 0 ends with all 8 totals.
// =====================================================================
typedef float v2f_t __attribute__((ext_vector_type(2)));
typedef float v8f_t __attribute__((ext_vector_type(8)));

__device__ __forceinline__ float compose_loss(const float* S) {
  float clrecall = (S[0] + 1e-12f) / (S[1] + 1e-12f);
  float clacc    = (S[2] + 1e-12f) / (S[3] + 1e-12f);
  float cldice   = 2.0f * clrecall * clacc / (clrecall + clacc);
  float cldice_loss = 1.0f - cldice;
  float dice_loss = 1.0f - (2.0f * S[4] + 1.0f) / (S[5] + S[6] + 1.0f);
  float mix = 0.5f * cldice_loss + 0.5f * dice_loss;
  float focal = S[7] / (float)NV_;
  return mix + 0.2f * focal;  // SCALE_CLDICE * mix + SCALE_FOCAL * focal
}

__global__ __launch_bounds__(32) void finalize(const float* __restrict__ partials,
                                               float* __restrict__ out) {
#if __has_builtin(__builtin_amdgcn_wmma_f32_16x16x4_f32)
  const int lane = threadIdx.x;
  const int row  = lane & 15;          // A-matrix row = quantity index
  const int kb   = (lane >> 4) << 1;   // lanes 16-31 supply K=2,3
  const float sc = (row < 8) ? 1.0f : 0.0f;
  const float* rp = partials + (size_t)((row < 8) ? row : 0) * NBLK;
  v8f_t acc = {0.f, 0.f, 0.f, 0.f, 0.f, 0.f, 0.f, 0.f};
  const v2f_t ones = {1.0f, 1.0f};
  for (int c = 0; c < NBLK; c += 4) {
    v2f_t a;
    a.x = rp[c + kb] * sc;
    a.y = rp[c + kb + 1] * sc;
    acc = __builtin_amdgcn_wmma_f32_16x16x4_f32(false, a, false, ones,
                                                (short)0, acc, false, false);
  }
  if (lane == 0) {
    float S[8];
#pragma unroll
    for (int j = 0; j < 8; ++j) S[j] = acc[j];  // D[M=j, N=0]
    out[0] = compose_loss(S);
  }
#else
  if (threadIdx.x == 0) {
    float S[8];
    for (int j = 0; j < 8; ++j) {
      float t = 0.f;
      for (int i = 0; i < NBLK; ++i) t += partials[(size_t)j * NBLK + i];
      S[j] = t;
    }
    out[0] = compose_loss(S);
  }
#endif
}

// =====================================================================
// Host launcher. Workspace layout (floats):
//   R0 [NV_] | R1 [NV_] | R2 [NV_] | partials [8*NBLK]
// ~108 MB total; with the two 37.7MB inputs the working set (~190MB)
// essentially fits the 192MB L2, so the iteration chain stays L2-resident.
// =====================================================================
extern "C" void kernel_launch(void* const* d_in, const int* in_sizes, int n_in,
                              void* d_out, int out_size, void* d_ws, size_t ws_size,
                              hipStream_t stream) {
  const float* pred   = (const float*)d_in[0];
  const float* target = (const float*)d_in[1];
  float* ws = (float*)d_ws;
  float* R0 = ws;
  float* R1 = ws + NV_;
  float* R2 = ws + 2 * NV_;
  float* part = ws + 3 * NV_;

  dim3 sgrid(WW / 16, HH / 16, (DD / 16) * 2);  // 12 x 12 x 16
  dim3 sblk(256);

  // pred skeleton: pred -> R0 -> R1 -> R0 -> R1 -> R0   (pskel = R0)
  {
    const float* s = pred;
    float* bufs[2] = {R0, R1};
    for (int it = 0; it < 5; ++it) {
      float* d = bufs[it & 1];
      skel_iter<<<sgrid, sblk, 0, stream>>>(s, d);
      s = d;
    }
  }
  // target skeleton: target -> R1 -> R2 -> R1 -> R2 -> R1   (tskel = R1)
  {
    const float* s = target;
    float* bufs[2] = {R1, R2};
    for (int it = 0; it < 5; ++it) {
      float* d = bufs[it & 1];
      skel_iter<<<sgrid, sblk, 0, stream>>>(s, d);
      s = d;
    }
  }

  reduce8<<<NBLK, 256, 0, stream>>>(pred, target, /*tskel=*/R1, /*pskel=*/R0, part);
  finalize<<<1, 32, 0, stream>>>(part, (float*)d_out);
}